// CorrelationLayer_52501680226681
// MI455X (gfx1250) — compile-verified
//
#include <hip/hip_runtime.h>

// FlowNet correlation on gfx1250 via V_WMMA_F32_16X16X4_F32 (fp32-exact).
// corr[b, di*9+j, h, w0+m] = sum_c x1[b,c,h,w0+m] * x2pad[b,c,h+di-4,w0+m+j-4]
// Per 16-pixel strip: A = x1[16 px x 64 ch]; B = x2 window [64 ch x 24 px],
// covered by two N=16 WMMA tiles (combined column index n = m + j, 0..23).

typedef __attribute__((ext_vector_type(2))) float v2f;
typedef __attribute__((ext_vector_type(8))) float v8f;

#define BB 8
#define CC 64
#define HH 192
#define WW 128
#define MD 4
#define DD 9            // 2*MD+1
#define ND 81           // DD*DD
#define HWSZ (HH * WW)
#define WSTRIPS (WW / 16)   // 8
#define WAVES 8             // waves per block

__device__ __forceinline__ v8f wmma4(v2f a, v2f b, v8f c) {
  return __builtin_amdgcn_wmma_f32_16x16x4_f32(false, a, false, b, (short)0, c,
                                               false, false);
}

// One displacement row: stage all B fragments (branchless, clamped addresses),
// then run the 32 accumulating WMMAs. EDGE strips zero out-of-range columns
// via selects (v_cndmask), interior strips are pure straight-line loads.
template <bool EDGE>
__device__ __forceinline__ void accum_row(const float* __restrict__ x2,
                                          const v2f a[16], long long rowoff,
                                          int xc0, int xc1, bool ok0, bool ok1,
                                          v8f& acc0, v8f& acc1) {
  v2f b0[16], b1[16];
#pragma unroll
  for (int kc = 0; kc < 16; ++kc) {
    const long long coff = rowoff + (long long)(kc * 4) * HWSZ;
    b0[kc].x = x2[coff + xc0];
    b0[kc].y = x2[coff + HWSZ + xc0];
    b1[kc].x = x2[coff + xc1];
    b1[kc].y = x2[coff + HWSZ + xc1];
  }
  if (EDGE) {
#pragma unroll
    for (int kc = 0; kc < 16; ++kc) {
      b0[kc].x = ok0 ? b0[kc].x : 0.f;
      b0[kc].y = ok0 ? b0[kc].y : 0.f;
      b1[kc].x = ok1 ? b1[kc].x : 0.f;
      b1[kc].y = ok1 ? b1[kc].y : 0.f;
    }
  }
#pragma unroll
  for (int kc = 0; kc < 16; ++kc) {
    acc0 = wmma4(a[kc], b0[kc], acc0);
    acc1 = wmma4(a[kc], b1[kc], acc1);
  }
}

__global__ __launch_bounds__(256) void corr_wmma_f32(
    const float* __restrict__ x1, const float* __restrict__ x2,
    float* __restrict__ out) {
  // Per-wave staging tile: 16 rows x 32 cols, padded stride 33 (conflict-free).
  __shared__ float lds[WAVES][16 * 33];

  const int lane = (int)(threadIdx.x & 31);
  const int wave = (int)(threadIdx.x >> 5);
  const int m  = lane & 15;   // pixel / column index within 16
  const int kh = lane >> 4;   // which K-half of the 16x4 / 4x16 fragment

  // Strip decode: tiles ordered (b, h, wstrip)
  const int tile = (int)blockIdx.x * WAVES + wave;
  const int ws = tile & (WSTRIPS - 1);
  const int t2 = tile >> 3;
  const int h  = t2 % HH;
  const int b  = t2 / HH;
  const int w0 = ws * 16;

  // ---- Preload A: x1 strip, 16 k-chunks of 16x4 fp32 ----
  // ISA layout: lane holds M = lane&15; VGPR v holds K = 2*(lane>>4) + v.
  v2f a[16];
  const size_t x1off = ((size_t)b * CC) * HWSZ + (size_t)h * WW + (size_t)(w0 + m);
#pragma unroll
  for (int kc = 0; kc < 16; ++kc) {
    const int c0 = kc * 4 + kh * 2;
    a[kc].x = x1[x1off + (size_t)c0 * HWSZ];
    a[kc].y = x1[x1off + (size_t)(c0 + 1) * HWSZ];
  }

  float* t = lds[wave];
  const bool edge = (ws == 0) || (ws == WSTRIPS - 1);   // wave-uniform
  const int xb0 = w0 - MD + m;    // block0: x2 columns w0-4 .. w0+11
  const int xb1 = w0 + 12 + m;    // block1: x2 columns w0+12 .. w0+27
  const int xc0 = min(max(xb0, 0), WW - 1);             // clamped (legal) addr
  const int xc1 = min(xb1, WW - 1);
  const bool ok0 = (xb0 >= 0) && (xb0 < WW);
  const bool ok1 = (xb1 < WW);
  // Fold batch base and the lane-half channel offset into one row base.
  const long long x2base =
      (long long)b * CC * HWSZ + (long long)(kh * 2) * HWSZ;

  for (int di = 0; di < DD; ++di) {
    const int hy = h + di - MD;
    const size_t obase =
        (((size_t)b * ND + (size_t)di * DD) * HH + (size_t)h) * WW +
        (size_t)(w0 + m);

    if (hy < 0 || hy >= HH) {      // vertical zero-padding: uniform fast path
#pragma unroll
      for (int j = kh; j < DD; j += 2)
        out[obase + (size_t)j * HWSZ] = 0.f;
      continue;
    }

    const long long rowoff = x2base + (long long)hy * WW;

    // Prefetch next displacement row (covers its 128B lines) so the fetch is
    // in flight across this iteration's compute + extraction fences.
    if (hy + 1 < HH) {
#pragma unroll
      for (int kc = 0; kc < 16; kc += 2)
        __builtin_prefetch(
            &x2[rowoff + (long long)WW + (long long)(kc * 4) * HWSZ + xc0],
            0, 0);
    }

    v8f acc0 = {};
    v8f acc1 = {};
    if (edge)
      accum_row<true>(x2, a, rowoff, xc0, xc1, ok0, ok1, acc0, acc1);
    else
      accum_row<false>(x2, a, rowoff, xc0, xc1, ok0, ok1, acc0, acc1);

    // ---- Stage D tiles to LDS (lane holds N=lane&15, VGPR r holds M=r+8*kh) ----
    asm volatile("" ::: "memory");
#pragma unroll
    for (int r = 0; r < 8; ++r) {
      const int mm = r + kh * 8;
      t[mm * 33 + m]      = acc0[r];   // combined col n = m        (block0)
      t[mm * 33 + 16 + m] = acc1[r];   // combined col n = 16 + m   (block1)
    }
    // Same-wave LDS is in-order; wait + compiler fence before cross-lane reads.
    asm volatile("s_wait_dscnt 0" ::: "memory");

    // ---- Extract band: out(m, j) = tile[m, m+j]; coalesced 16-float rows ----
#pragma unroll
    for (int j = kh; j < DD; j += 2) {     // half-wave j=0,2,4,6,8 ; half 1,3,5,7
      out[obase + (size_t)j * HWSZ] = t[m * 34 + j];   // m*33 + (m+j)
    }
    asm volatile("s_wait_dscnt 0" ::: "memory");  // reads retire before next stores
  }
}

extern "C" void kernel_launch(void* const* d_in, const int* in_sizes, int n_in,
                              void* d_out, int out_size, void* d_ws, size_t ws_size,
                              hipStream_t stream) {
  (void)in_sizes; (void)n_in; (void)d_ws; (void)ws_size; (void)out_size;
  const float* x1 = (const float*)d_in[0];
  const float* x2 = (const float*)d_in[1];
  float* out = (float*)d_out;

  const int total_tiles = BB * HH * WSTRIPS;          // 12288 strips
  const int blocks = total_tiles / WAVES;             // 1536
  corr_wmma_f32<<<blocks, 256, 0, stream>>>(x1, x2, out);
}